// GraphConvolution_71554155151512
// MI455X (gfx1250) — compile-verified
//
#include <hip/hip_runtime.h>
#include <stdint.h>

#define N_NODES 12288
#define IN_F 64
#define OUT_F 32
#define BK 64
#define A_PAD 68              // 16B-aligned float4 rows; A-frag banks spread as 4*m
#define B_PAD 36              // 16B-aligned float4 rows; B rows bank-rotate by 4
#define ABUF (16 * A_PAD)     // floats per A buffer (1088)
#define BBUF (BK * B_PAD)     // floats per B buffer (2304)
#define BUFSZ (ABUF + BBUF)   // 3392 floats per double-buffer slot
#define RED_OFF (2 * BUFSZ)   // reduction scratch after the two slots

typedef __attribute__((ext_vector_type(2))) float v2f;
typedef __attribute__((ext_vector_type(8))) float v8f;

// ---------- Kernel 1: dinv[j] = 1 / (1 + sum_k adj[j][k]) ----------
__global__ void __launch_bounds__(256) gcn_rowsum_kernel(const float* __restrict__ adj,
                                                         float* __restrict__ dinv) {
    __shared__ float red[256];
    const int row = blockIdx.x;
    const int t = threadIdx.x;
    const float4* rp = (const float4*)(adj + (size_t)row * N_NODES);
    float s = 0.f;
#pragma unroll
    for (int i = 0; i < (N_NODES / 4) / 256; ++i) {
        float4 v = rp[t + i * 256];
        s += v.x + v.y + v.z + v.w;
    }
    red[t] = s;
    __syncthreads();
    for (int off = 128; off > 0; off >>= 1) {
        if (t < off) red[t] += red[t + off];
        __syncthreads();
    }
    if (t == 0) {
        float d = red[0] + 1.0f;                   // +1 from identity diagonal
        dinv[row] = (d > 0.f) ? (1.0f / d) : 0.0f; // matches isinf->0 masking
    }
}

// ---------- Kernel 2: S[j,k] = dinv[j] * (X[j,:] . W[k,:] + b[k]) ----------
__global__ void __launch_bounds__(256) gcn_support_kernel(const float* __restrict__ X,
                                                          const float* __restrict__ W,
                                                          const float* __restrict__ bias,
                                                          const float* __restrict__ dinv,
                                                          float* __restrict__ S) {
    __shared__ float Ws[OUT_F * IN_F];
    __shared__ float bs[OUT_F];
    const int t = threadIdx.x;
    for (int i = t; i < OUT_F * IN_F; i += 256) Ws[i] = W[i];
    if (t < OUT_F) bs[t] = bias[t];
    __syncthreads();
    const int idx = blockIdx.x * 256 + t;
    if (idx >= N_NODES * OUT_F) return;
    const int j = idx >> 5;
    const int k = idx & 31;
    const float* xr = X + j * IN_F;
    float acc = bs[k];
#pragma unroll
    for (int c = 0; c < IN_F; ++c) acc += xr[c] * Ws[k * IN_F + c];
    S[idx] = dinv[j] * acc;
}

// ---------- Kernel 3: out = relu(A @ S + S) via V_WMMA_F32_16X16X4_F32 ----------
// 128 threads = 4 wave32s per block; block owns 16 output rows x 32 cols.
// wave = (ntile, kpart). Double-buffered LDS fed by GLOBAL_LOAD_ASYNC_TO_LDS_B128.

__device__ __forceinline__ void async_copy16(uint32_t lds_byte_off, const float* gptr) {
    // CDNA5 async DMA: global memory -> LDS, tracked by ASYNCcnt (no VGPR round-trip)
    asm volatile("global_load_async_to_lds_b128 %0, %1, off"
                 :: "v"(lds_byte_off), "v"(gptr)
                 : "memory");
}

__device__ __forceinline__ void wait_async_all() {
    asm volatile("s_wait_asynccnt 0x0" ::: "memory");
}

__global__ void __launch_bounds__(128) gcn_gemm_kernel(const float* __restrict__ adj,
                                                       const float* __restrict__ S,
                                                       float* __restrict__ out) {
    __shared__ float lds[2 * BUFSZ + 2 * 8 * 32];

    const int t = threadIdx.x;
    const int wave = t >> 5;
    const int lane = t & 31;
    const int ntile = wave & 1;   // 16-col half
    const int kpart = wave >> 1;  // K sub-range of each chunk
    const int g = lane >> 4;      // half-wave group
    const int mn = lane & 15;     // M for A / N for B,C

    const int row0 = blockIdx.x * 16;

    // Per-thread staging assignments (constant across chunks)
    const int aIdx0 = t, aIdx1 = t + 128;              // 256 float4s of A
    const int ar0 = aIdx0 >> 4, ac0 = aIdx0 & 15;
    const int ar1 = aIdx1 >> 4, ac1 = aIdx1 & 15;

    v8f c = {};

    // Prologue: stage chunk 0 into slot 0
    {
        const uint32_t aB = 0, bB = ABUF * 4;
        async_copy16(aB + (uint32_t)(ar0 * A_PAD + ac0 * 4) * 4,
                     adj + (size_t)(row0 + ar0) * N_NODES + ac0 * 4);
        async_copy16(aB + (uint32_t)(ar1 * A_PAD + ac1 * 4) * 4,
                     adj + (size_t)(row0 + ar1) * N_NODES + ac1 * 4);
#pragma unroll
        for (int i = 0; i < 4; ++i) {
            int idx = t + i * 128;
            int r = idx >> 3, c4 = idx & 7;
            async_copy16(bB + (uint32_t)(r * B_PAD + c4 * 4) * 4,
                         S + (size_t)r * OUT_F + c4 * 4);
        }
    }
    wait_async_all();
    __syncthreads();

    const int nchunks = N_NODES / BK;  // 192
    for (int ci = 0; ci < nchunks; ++ci) {
        const int cur = ci & 1;
        const int nxt = cur ^ 1;

        // Prefetch chunk ci+1 into the other slot (fire-and-forget)
        if (ci + 1 < nchunks) {
            const int k1 = (ci + 1) * BK;
            const uint32_t aB = (uint32_t)(nxt * BUFSZ) * 4;
            const uint32_t bB = aB + ABUF * 4;
            async_copy16(aB + (uint32_t)(ar0 * A_PAD + ac0 * 4) * 4,
                         adj + (size_t)(row0 + ar0) * N_NODES + k1 + ac0 * 4);
            async_copy16(aB + (uint32_t)(ar1 * A_PAD + ac1 * 4) * 4,
                         adj + (size_t)(row0 + ar1) * N_NODES + k1 + ac1 * 4);
#pragma unroll
            for (int i = 0; i < 4; ++i) {
                int idx = t + i * 128;
                int r = idx >> 3, c4 = idx & 7;
                async_copy16(bB + (uint32_t)(r * B_PAD + c4 * 4) * 4,
                             S + (size_t)(k1 + r) * OUT_F + c4 * 4);
            }
        }

        // Compute on current slot: gather all fragments, then 8 back-to-back WMMAs
        const float* Al = &lds[cur * BUFSZ];
        const float* Bl = Al + ABUF;
        v2f af[8], bf[8];
#pragma unroll
        for (int kk = 0; kk < 8; ++kk) {
            const int k = kpart * 32 + kk * 4;
            af[kk] = *(const v2f*)(&Al[mn * A_PAD + k + 2 * g]);
            bf[kk].x = Bl[(k + 2 * g) * B_PAD + ntile * 16 + mn];
            bf[kk].y = Bl[(k + 2 * g + 1) * B_PAD + ntile * 16 + mn];
        }
#pragma unroll
        for (int kk = 0; kk < 8; ++kk) {
            c = __builtin_amdgcn_wmma_f32_16x16x4_f32(false, af[kk], false, bf[kk],
                                                      (short)0, c, false, false);
        }

        wait_async_all();   // prefetch landed
        __syncthreads();    // all waves' prefetches visible; slot 'cur' free to reuse
    }

    // Combine K-halves: kpart==1 publishes, kpart==0 reduces + epilogue
    float* Red = &lds[RED_OFF];
    if (kpart == 1) {
#pragma unroll
        for (int i = 0; i < 8; ++i) Red[ntile * 256 + i * 32 + lane] = c[i];
    }
    __syncthreads();
    if (kpart == 0) {
#pragma unroll
        for (int i = 0; i < 8; ++i) {
            const int m = i + 8 * g;           // C layout: VGPR i -> M=i (+8 hi half)
            const int col = ntile * 16 + mn;
            const int row = row0 + m;
            float v = c[i] + Red[ntile * 256 + i * 32 + lane];
            v += S[row * OUT_F + col];         // identity (A + I) contribution
            out[row * OUT_F + col] = fmaxf(v, 0.0f);
        }
    }
}

extern "C" void kernel_launch(void* const* d_in, const int* in_sizes, int n_in,
                              void* d_out, int out_size, void* d_ws, size_t ws_size,
                              hipStream_t stream) {
    const float* X   = (const float*)d_in[0];  // node_features (N x 64)
    const float* adj = (const float*)d_in[1];  // adj_matrix   (N x N)
    const float* W   = (const float*)d_in[2];  // W            (32 x 64)
    const float* b   = (const float*)d_in[3];  // b            (32)
    float* out = (float*)d_out;

    float* dinv = (float*)d_ws;                // N floats
    float* Smat = dinv + N_NODES;              // N x 32 floats

    gcn_rowsum_kernel<<<N_NODES, 256, 0, stream>>>(adj, dinv);
    gcn_support_kernel<<<(N_NODES * OUT_F + 255) / 256, 256, 0, stream>>>(X, W, b, dinv, Smat);
    gcn_gemm_kernel<<<N_NODES / 16, 128, 0, stream>>>(adj, Smat, out);
}